// SPDRectified_53060025974941
// MI455X (gfx1250) — compile-verified
//
#include <hip/hip_runtime.h>

#define EPS 1e-4f
#define NMAT 32
#define LD 36                 // LDS row stride (floats): float4 16B / float2 8B aligned
#define WAVES_PER_BLOCK 4
#define NS_ITERS 14

typedef float v2f __attribute__((ext_vector_type(2)));
typedef float v8f __attribute__((ext_vector_type(8)));

__device__ __forceinline__ v8f wmma4(v2f a, v2f b, v8f c) {
  // D = A(16x4,f32) * B(4x16,f32) + C(16x16,f32)
  return __builtin_amdgcn_wmma_f32_16x16x4_f32(false, a, false, b, (short)0, c,
                                               false, false);
}

// acc(2x2 tiles) += Asrc * Bsrc, where Bsrc is SYMMETRIC: B-fragments are loaded
// as contiguous float2 rows (B[k][n] == B[n][k]), same shape as A-fragments.
__device__ __forceinline__ void mm32_sym(const float* Asrc, const float* Bsrc,
                                         v8f& c00, v8f& c01, v8f& c10, v8f& c11,
                                         int lo, int hi) {
#pragma unroll
  for (int kb = 0; kb < NMAT; kb += 4) {
    const int kc = kb + hi * 2;
    v2f a0 = *(const v2f*)&Asrc[lo * LD + kc];
    v2f a1 = *(const v2f*)&Asrc[(16 + lo) * LD + kc];
    v2f b0 = *(const v2f*)&Bsrc[lo * LD + kc];        // = B[kc..kc+1][lo]
    v2f b1 = *(const v2f*)&Bsrc[(16 + lo) * LD + kc]; // = B[kc..kc+1][16+lo]
    c00 = wmma4(a0, b0, c00);
    c01 = wmma4(a0, b1, c01);
    c10 = wmma4(a1, b0, c10);
    c11 = wmma4(a1, b1, c11);
  }
}

// acc = Xsrc * Xsrc for SYMMETRIC Xsrc: A-frags double as B-frags (2 loads/k-step).
__device__ __forceinline__ void mm32_square(const float* Xsrc,
                                            v8f& c00, v8f& c01, v8f& c10, v8f& c11,
                                            int lo, int hi) {
#pragma unroll
  for (int kb = 0; kb < NMAT; kb += 4) {
    const int kc = kb + hi * 2;
    v2f a0 = *(const v2f*)&Xsrc[lo * LD + kc];
    v2f a1 = *(const v2f*)&Xsrc[(16 + lo) * LD + kc];
    c00 = wmma4(a0, a0, c00);
    c01 = wmma4(a0, a1, c01);
    c10 = wmma4(a1, a0, c10);
    c11 = wmma4(a1, a1, c11);
  }
}

__global__ __launch_bounds__(WAVES_PER_BLOCK * 32)
void SPDRectified_ns_wmma(const float* __restrict__ A, float* __restrict__ out,
                          int batch) {
  __shared__ float lds[WAVES_PER_BLOCK][3][NMAT * LD];
  const int lane = threadIdx.x & 31;
  const int wid  = threadIdx.x >> 5;
  const int b    = blockIdx.x * WAVES_PER_BLOCK + wid;
  if (b >= batch) return;               // uniform per wave
  const int lo = lane & 15;
  const int hi = lane >> 4;

  float* Ms = lds[wid][0];   // M = A - eps*I
  float* Xs = lds[wid][1];   // Newton-Schulz iterate
  float* Ts = lds[wid][2];   // 3I - X^2

  const float* Ab = A + (size_t)b * (NMAT * NMAT);

  // ---- Load A (coalesced float4), M = A - eps*I -> LDS, Frobenius^2; keep regs ----
  float4 r[8];
  float ss = 0.f;
#pragma unroll
  for (int i = 0; i < 8; ++i) {
    const int idx = i * 128 + lane * 4;          // flat element index, 0..1023
    r[i] = *(const float4*)(Ab + idx);
    const int row = idx >> 5, col = idx & 31;
    const int d = row - col;
    if (d >= 0 && d < 4) ((float*)&r[i])[d] -= EPS;
    *(float4*)&Ms[row * LD + col] = r[i];        // (row*36+col)*4 is 16B aligned
    ss += r[i].x * r[i].x + r[i].y * r[i].y + r[i].z * r[i].z + r[i].w * r[i].w;
  }
#pragma unroll
  for (int off = 16; off; off >>= 1) ss += __shfl_xor(ss, off, 32);
  const float inv_c = rsqrtf(ss);                // 1/||M||_F bounds spectral norm

  // ---- X0 = M / ||M||_F, written straight from registers ----
#pragma unroll
  for (int i = 0; i < 8; ++i) {
    const int idx = i * 128 + lane * 4;
    const int row = idx >> 5, col = idx & 31;
    float4 v = make_float4(r[i].x * inv_c, r[i].y * inv_c,
                           r[i].z * inv_c, r[i].w * inv_c);
    *(float4*)&Xs[row * LD + col] = v;
  }

  const v8f vz = {0.f, 0.f, 0.f, 0.f, 0.f, 0.f, 0.f, 0.f};

  // ---- Newton-Schulz: X <- 0.5 * X * (3I - X^2), converges to sign(M) ----
#pragma unroll 1
  for (int it = 0; it < NS_ITERS; ++it) {
    v8f y00 = vz, y01 = vz, y10 = vz, y11 = vz;
    mm32_square(Xs, y00, y01, y10, y11, lo, hi); // Y = X*X (symmetric X)
#pragma unroll
    for (int v = 0; v < 8; ++v) {                // T = 3I - Y  (C/D layout -> LDS)
      const int rr = hi * 8 + v;
      const float dg = (lo == rr) ? 3.0f : 0.0f; // diagonal in (0,0)/(1,1) tiles
      Ts[rr * LD + lo]             = dg - y00[v];
      Ts[rr * LD + 16 + lo]        = -y01[v];
      Ts[(16 + rr) * LD + lo]      = -y10[v];
      Ts[(16 + rr) * LD + 16 + lo] = dg - y11[v];
    }
    v8f z00 = vz, z01 = vz, z10 = vz, z11 = vz;
    mm32_sym(Xs, Ts, z00, z01, z10, z11, lo, hi); // Z = X*T (T symmetric)
#pragma unroll
    for (int v = 0; v < 8; ++v) {                // X = 0.5*Z
      const int rr = hi * 8 + v;
      Xs[rr * LD + lo]             = 0.5f * z00[v];
      Xs[rr * LD + 16 + lo]        = 0.5f * z01[v];
      Xs[(16 + rr) * LD + lo]      = 0.5f * z10[v];
      Xs[(16 + rr) * LD + 16 + lo] = 0.5f * z11[v];
    }
  }

  // ---- P = M * sign(M) = |M| ; out = 0.5*M + 0.5*P + eps*I ----
  v8f p00 = vz, p01 = vz, p10 = vz, p11 = vz;
  mm32_sym(Ms, Xs, p00, p01, p10, p11, lo, hi);  // sign(M) symmetric

  float* Ob = out + (size_t)b * (NMAT * NMAT);
#pragma unroll
  for (int v = 0; v < 8; ++v) {
    const int rr = hi * 8 + v;
    Ob[rr * 32 + lo] =
        0.5f * (Ms[rr * LD + lo] + p00[v]) + ((lo == rr) ? EPS : 0.f);
    Ob[rr * 32 + 16 + lo] = 0.5f * (Ms[rr * LD + 16 + lo] + p01[v]);
    Ob[(16 + rr) * 32 + lo] = 0.5f * (Ms[(16 + rr) * LD + lo] + p10[v]);
    Ob[(16 + rr) * 32 + 16 + lo] =
        0.5f * (Ms[(16 + rr) * LD + 16 + lo] + p11[v]) + ((lo == rr) ? EPS : 0.f);
  }
}

extern "C" void kernel_launch(void* const* d_in, const int* in_sizes, int n_in,
                              void* d_out, int out_size, void* d_ws, size_t ws_size,
                              hipStream_t stream) {
  const float* A = (const float*)d_in[0];
  float* out = (float*)d_out;
  const int batch = in_sizes[0] / (NMAT * NMAT);   // 65536
  const int blocks = (batch + WAVES_PER_BLOCK - 1) / WAVES_PER_BLOCK;
  SPDRectified_ns_wmma<<<blocks, WAVES_PER_BLOCK * 32, 0, stream>>>(A, out, batch);
}